// SimpleHeteroSAGE_51711406244226
// MI455X (gfx1250) — compile-verified
//
#include <hip/hip_runtime.h>

#define NUQ 200000
#define NIQ 200000
#define NEDGE 500000
#define DIM 128
#define HF 128
#define SLOPE 0.01f

typedef __attribute__((ext_vector_type(2))) float v2f;
typedef __attribute__((ext_vector_type(8))) float v8f;

__global__ void zero_kernel(float* __restrict__ p, int n) {
    int i = blockIdx.x * blockDim.x + threadIdx.x;
    if (i < n) p[i] = 0.0f;
}

__global__ void deg_kernel(const int* __restrict__ dst, float* __restrict__ deg, int e) {
    int i = blockIdx.x * blockDim.x + threadIdx.x;
    if (i < e) atomicAdd(&deg[dst[i]], 1.0f);
}

__global__ void rdeg_kernel(float* __restrict__ deg, int n) {
    int i = blockIdx.x * blockDim.x + threadIdx.x;
    if (i < n) deg[i] = 1.0f / fmaxf(deg[i], 1.0f);
}

// one thread per (edge, feature): coalesced gather of 128 floats per edge,
// coalesced fp32 atomic add into the aggregation buffer (lives in 192MB L2)
__global__ void scatter_kernel(const float* __restrict__ x,
                               const int* __restrict__ src,
                               const int* __restrict__ dst,
                               float* __restrict__ agg, int e) {
    int idx = blockIdx.x * blockDim.x + threadIdx.x;
    int eid = idx >> 7;
    int f = idx & 127;
    if (eid < e) {
        int s = src[eid];
        int d = dst[eid];
        atomicAdd(&agg[d * DIM + f], x[s * DIM + f]);
    }
}

// Fused SAGE layer:  out = act( x_dst @ Wself + (agg * rdeg) @ Wneigh + b )
// Block: 16-row panel, 8 waves x 16-col tiles, fp32 WMMA 16x16x4, K=128.
__launch_bounds__(256)
__global__ void sage_gemm_kernel(const float* __restrict__ xdst,
                                 const float* __restrict__ agg,
                                 const float* __restrict__ rdeg,
                                 const float* __restrict__ Wself,
                                 const float* __restrict__ Wneigh,
                                 const float* __restrict__ bias,
                                 float* __restrict__ out,
                                 int apply_relu) {
    __shared__ float xs[16 * DIM];   // x_dst panel
    __shared__ float ns[16 * DIM];   // mean (agg * rdeg) panel

    const int row0 = blockIdx.x * 16;
    const int tid = threadIdx.x;

    // cooperative stage of both A panels into LDS (8 elems per thread each)
    #pragma unroll
    for (int i = tid; i < 16 * DIM; i += 256) {
        int r = i >> 7;
        int c = i & 127;
        float rd = rdeg[row0 + r];
        xs[i] = xdst[(row0 + r) * DIM + c];
        ns[i] = agg[(row0 + r) * DIM + c] * rd;
    }
    __syncthreads();

    const int wave = tid >> 5;      // 0..7 -> column tile
    const int lane = tid & 31;
    const int col0 = wave * 16;
    const int mn = lane & 15;       // A row (M) == B/C column (N) for this lane
    const int kh = lane >> 4;       // 0: K=k0,k0+1   1: K=k0+2,k0+3

    v8f acc = {};
    for (int k0 = 0; k0 < DIM; k0 += 4) {
        const int ka = k0 + 2 * kh;
        // A 16x4 fp32: VGPR0 = K even slot, VGPR1 = next K (per ISA layout)
        v2f aS = *(const v2f*)&xs[mn * DIM + ka];
        v2f aN = *(const v2f*)&ns[mn * DIM + ka];
        // B 4x16 fp32: N striped across lanes within a VGPR
        v2f bS, bN;
        bS.x = Wself[(ka) * HF + col0 + mn];
        bS.y = Wself[(ka + 1) * HF + col0 + mn];
        bN.x = Wneigh[(ka) * HF + col0 + mn];
        bN.y = Wneigh[(ka + 1) * HF + col0 + mn];
        acc = __builtin_amdgcn_wmma_f32_16x16x4_f32(
            false, aS, false, bS, (short)0, acc, false, false);
        acc = __builtin_amdgcn_wmma_f32_16x16x4_f32(
            false, aN, false, bN, (short)0, acc, false, false);
    }

    const float bv = bias[col0 + mn];
    #pragma unroll
    for (int r = 0; r < 8; ++r) {
        float v = acc[r] + bv;
        if (apply_relu) v = (v > 0.0f) ? v : SLOPE * v;
        // C/D layout: VGPR r -> M=r (lanes 0-15), M=r+8 (lanes 16-31)
        out[(row0 + r + kh * 8) * HF + col0 + mn] = v;
    }
}

static inline int cdiv(int a, int b) { return (a + b - 1) / b; }

extern "C" void kernel_launch(void* const* d_in, const int* in_sizes, int n_in,
                              void* d_out, int out_size, void* d_ws, size_t ws_size,
                              hipStream_t stream) {
    const float* x_user      = (const float*)d_in[0];
    const float* x_item      = (const float*)d_in[1];
    const float* w1_self_uc  = (const float*)d_in[2];
    const float* w1_neigh_uc = (const float*)d_in[3];
    const float* b1_uc       = (const float*)d_in[4];
    const float* w1_self_iu  = (const float*)d_in[5];
    const float* w1_neigh_iu = (const float*)d_in[6];
    const float* b1_iu       = (const float*)d_in[7];
    const float* w2_self_uc  = (const float*)d_in[8];
    const float* w2_neigh_uc = (const float*)d_in[9];
    const float* b2_uc       = (const float*)d_in[10];
    const float* w2_self_iu  = (const float*)d_in[11];
    const float* w2_neigh_iu = (const float*)d_in[12];
    const float* b2_iu       = (const float*)d_in[13];
    const int*   src_uc      = (const int*)d_in[14];
    const int*   dst_uc      = (const int*)d_in[15];
    const int*   src_iu      = (const int*)d_in[16];
    const int*   dst_iu      = (const int*)d_in[17];

    float* ws     = (float*)d_ws;
    float* agg_i  = ws;                                // NI*128
    float* agg_u  = agg_i + (size_t)NIQ * DIM;         // NU*128
    float* h1_i   = agg_u + (size_t)NUQ * DIM;         // NI*128
    float* h1_u   = h1_i + (size_t)NIQ * DIM;          // NU*128
    float* rdeg_i = h1_u + (size_t)NUQ * DIM;          // NI
    float* rdeg_u = rdeg_i + NIQ;                      // NU

    float* h2_user = (float*)d_out;                    // [NU,128]
    float* h2_item = (float*)d_out + (size_t)NUQ * HF; // [NI,128]

    const dim3 B(256);

    // ---- degrees (shared by both layers) ----
    zero_kernel<<<cdiv(NIQ, 256), B, 0, stream>>>(rdeg_i, NIQ);
    zero_kernel<<<cdiv(NUQ, 256), B, 0, stream>>>(rdeg_u, NUQ);
    deg_kernel<<<cdiv(NEDGE, 256), B, 0, stream>>>(dst_uc, rdeg_i, NEDGE);
    deg_kernel<<<cdiv(NEDGE, 256), B, 0, stream>>>(dst_iu, rdeg_u, NEDGE);
    rdeg_kernel<<<cdiv(NIQ, 256), B, 0, stream>>>(rdeg_i, NIQ);
    rdeg_kernel<<<cdiv(NUQ, 256), B, 0, stream>>>(rdeg_u, NUQ);

    // ---- layer 1 aggregation ----
    zero_kernel<<<cdiv(NIQ * DIM, 256), B, 0, stream>>>(agg_i, NIQ * DIM);
    zero_kernel<<<cdiv(NUQ * DIM, 256), B, 0, stream>>>(agg_u, NUQ * DIM);
    scatter_kernel<<<cdiv(NEDGE * DIM, 256), B, 0, stream>>>(x_user, src_uc, dst_uc, agg_i, NEDGE);
    scatter_kernel<<<cdiv(NEDGE * DIM, 256), B, 0, stream>>>(x_item, src_iu, dst_iu, agg_u, NEDGE);

    // ---- layer 1 fused GEMM + leaky relu ----
    sage_gemm_kernel<<<NIQ / 16, B, 0, stream>>>(x_item, agg_i, rdeg_i,
        w1_self_uc, w1_neigh_uc, b1_uc, h1_i, 1);
    sage_gemm_kernel<<<NUQ / 16, B, 0, stream>>>(x_user, agg_u, rdeg_u,
        w1_self_iu, w1_neigh_iu, b1_iu, h1_u, 1);

    // ---- layer 2 aggregation ----
    zero_kernel<<<cdiv(NIQ * DIM, 256), B, 0, stream>>>(agg_i, NIQ * DIM);
    zero_kernel<<<cdiv(NUQ * DIM, 256), B, 0, stream>>>(agg_u, NUQ * DIM);
    scatter_kernel<<<cdiv(NEDGE * DIM, 256), B, 0, stream>>>(h1_u, src_uc, dst_uc, agg_i, NEDGE);
    scatter_kernel<<<cdiv(NEDGE * DIM, 256), B, 0, stream>>>(h1_i, src_iu, dst_iu, agg_u, NEDGE);

    // ---- layer 2 fused GEMM (no activation), straight into d_out ----
    sage_gemm_kernel<<<NIQ / 16, B, 0, stream>>>(h1_i, agg_i, rdeg_i,
        w2_self_uc, w2_neigh_uc, b2_uc, h2_item, 0);
    sage_gemm_kernel<<<NUQ / 16, B, 0, stream>>>(h1_u, agg_u, rdeg_u,
        w2_self_iu, w2_neigh_iu, b2_iu, h2_user, 0);
}